// TNModel_non_hermitian_11295763988909
// MI455X (gfx1250) — compile-verified
//
#include <hip/hip_runtime.h>

// MPS/MPO problem constants (from the reference)
#define NSITE 32

typedef float v2f __attribute__((ext_vector_type(2)));
typedef float v8f __attribute__((ext_vector_type(8)));

// ---------------------------------------------------------------------------
// Generic complex GEMM, f32, via V_WMMA_F32_16X16X4_F32.
//   C[m,n] = sum_k (CONJA ? conj(A[m,k]) : A[m,k]) * B[k,n]
// m=(m1,m0), n=(n1,n0), k=(k1,k0) composite indices, each with its own element
// stride (permuted/reshaped tensor views need no explicit transpose).
// Complex storage is SoA (separate re/im planes, same strides).
//
// One wave computes a 32x64 complex tile = 2x4 WMMA sub-tiles -> 32 f32 WMMAs
// per K=4 step against 24 scalar loads. Hot-loop properties:
//  - rows/cols CLAMPED (out-of-range A rows only feed the same out-of-range C
//    rows, never stored) -> unconditional loads, no exec masking;
//  - K offsets maintained incrementally. If K0 divides the step (K0 in {1,2})
//    the wrap count is constant and folded into the per-step delta host-side
//    (K0eff disables the check); if K0 >= 4 at most one wrap per step occurs
//    and is handled with a branchless select -> no divergent loops;
//  - next fragments prefetched before the current accumulate (pipelined).
// K must be a multiple of 4 (true for every call below).
// ---------------------------------------------------------------------------
template<int CONJA>
__global__ void __launch_bounds__(32)
cgemm_wmma(int M1, int M0, int N1, int N0, int K,
           const float* __restrict__ Are, const float* __restrict__ Aim,
           long sa_m1, long sa_m0, long sa_k0, long adjA, long dA,
           const float* __restrict__ Bre, const float* __restrict__ Bim,
           long sb_k0, long adjB, long dB, long sb_n1, long sb_n0,
           int K0init, int K0eff,
           float* __restrict__ Cre, float* __restrict__ Cim,
           long sc_m1, long sc_m0, long sc_n1, long sc_n0)
{
  const int lane = threadIdx.x & 31;
  const int hf   = lane >> 4;     // 0: lanes 0-15, 1: lanes 16-31
  const int l15  = lane & 15;
  const int M = M1 * M0;
  const int N = N1 * N0;
  const int tM = blockIdx.y;      // 32-row tile
  const int tN = blockIdx.x;      // 64-col tile

  // ---- per-lane A-row / B-col base offsets (clamped; loop-invariant) ----
  long aOff[2], bOff[4];
#pragma unroll
  for (int i = 0; i < 2; ++i) {
    int r = tM * 32 + i * 16 + l15;  if (r > M - 1) r = M - 1;
    const int r1 = r / M0, r0i = r - r1 * M0;
    aOff[i] = (long)r1 * sa_m1 + (long)r0i * sa_m0;
  }
#pragma unroll
  for (int i = 0; i < 4; ++i) {
    int c = tN * 64 + i * 16 + l15;  if (c > N - 1) c = N - 1;
    const int c1 = c / N0, c0 = c - c1 * N0;
    bOff[i] = (long)c1 * sb_n1 + (long)c0 * sb_n0;
  }

  // ---- incremental K-offset state (two K slots: ka = kb+hf*2, kc = ka+1) --
  int  r0 = hf * 2,       r1 = hf * 2 + 1;
  long oa0 = (long)r0 * sa_k0, ob0 = (long)r0 * sb_k0;
  long oa1 = (long)r1 * sa_k0, ob1 = (long)r1 * sb_k0;
  while (r0 >= K0init) { r0 -= K0init; oa0 += adjA; ob0 += adjB; }  // cold init
  while (r1 >= K0init) { r1 -= K0init; oa1 += adjA; ob1 += adjB; }

  // fragments (current + prefetched next)
  v2f ar[2], ai[2], br[4], bi[4];
  v2f nar[2], nai[2], nbr[4], nbi[4];

  auto ldfrag = [&](v2f par[2], v2f pai[2], v2f pbr[4], v2f pbi[4]) {
#pragma unroll
    for (int i = 0; i < 2; ++i) {
      par[i].x = Are[aOff[i] + oa0];  par[i].y = Are[aOff[i] + oa1];
      pai[i].x = Aim[aOff[i] + oa0];  pai[i].y = Aim[aOff[i] + oa1];
    }
#pragma unroll
    for (int i = 0; i < 4; ++i) {
      pbr[i].x = Bre[bOff[i] + ob0];  pbr[i].y = Bre[bOff[i] + ob1];
      pbi[i].x = Bim[bOff[i] + ob0];  pbi[i].y = Bim[bOff[i] + ob1];
    }
  };
  auto step = [&]() {
    r0 += 4; oa0 += dA; ob0 += dB;
    { const bool w = (r0 >= K0eff);
      r0  -= w ? K0eff : 0;  oa0 += w ? adjA : 0;  ob0 += w ? adjB : 0; }
    r1 += 4; oa1 += dA; ob1 += dB;
    { const bool w = (r1 >= K0eff);
      r1  -= w ? K0eff : 0;  oa1 += w ? adjA : 0;  ob1 += w ? adjB : 0; }
  };

  // accumulators: 2x4 complex 16x16 tiles
  v8f cre[2][4], cim[2][4];
#pragma unroll
  for (int mi = 0; mi < 2; ++mi)
#pragma unroll
    for (int ni = 0; ni < 4; ++ni) {
      cre[mi][ni] = (v8f){0,0,0,0,0,0,0,0};
      cim[mi][ni] = (v8f){0,0,0,0,0,0,0,0};
    }

  auto accum = [&](const v2f xar[2], const v2f xai[2],
                   const v2f xbr[4], const v2f xbi[4]) {
#pragma unroll
    for (int mi = 0; mi < 2; ++mi) {
      v2f neg; neg.x = -xai[mi].x; neg.y = -xai[mi].y;
      // CONJA=0: Cre += ArBr - AiBi ; Cim += ArBi + AiBr
      // CONJA=1: Cre += ArBr + AiBi ; Cim += ArBi - AiBr
      const v2f aRe = CONJA ? xai[mi] : neg;
      const v2f aIm = CONJA ? neg : xai[mi];
#pragma unroll
      for (int ni = 0; ni < 4; ++ni) {
        cre[mi][ni] = __builtin_amdgcn_wmma_f32_16x16x4_f32(
            false, xar[mi], false, xbr[ni], (short)0, cre[mi][ni], false, false);
        cre[mi][ni] = __builtin_amdgcn_wmma_f32_16x16x4_f32(
            false, aRe,     false, xbi[ni], (short)0, cre[mi][ni], false, false);
        cim[mi][ni] = __builtin_amdgcn_wmma_f32_16x16x4_f32(
            false, xar[mi], false, xbi[ni], (short)0, cim[mi][ni], false, false);
        cim[mi][ni] = __builtin_amdgcn_wmma_f32_16x16x4_f32(
            false, aIm,     false, xbr[ni], (short)0, cim[mi][ni], false, false);
      }
    }
  };

  // ---- pipelined K loop ----
  ldfrag(ar, ai, br, bi);
  for (int kb = 4; kb < K; kb += 4) {
    step();
    ldfrag(nar, nai, nbr, nbi);
    accum(ar, ai, br, bi);
#pragma unroll
    for (int i = 0; i < 2; ++i) { ar[i] = nar[i]; ai[i] = nai[i]; }
#pragma unroll
    for (int i = 0; i < 4; ++i) { br[i] = nbr[i]; bi[i] = nbi[i]; }
  }
  accum(ar, ai, br, bi);

  // ---- store (guarded; outside hot loop) ----
#pragma unroll
  for (int ni = 0; ni < 4; ++ni) {
    const int ccol = tN * 64 + ni * 16 + l15;
    if (ccol >= N) continue;
    const int cn1 = ccol / N0, cn0 = ccol - cn1 * N0;
    const long cColOff = (long)cn1 * sc_n1 + (long)cn0 * sc_n0;
#pragma unroll
    for (int mi = 0; mi < 2; ++mi) {
#pragma unroll
      for (int i = 0; i < 8; ++i) {
        const int crow = tM * 32 + mi * 16 + hf * 8 + i;
        if (crow < M) {
          const int cm1 = crow / M0, cm0 = crow - cm1 * M0;
          const long off = (long)cm1 * sc_m1 + (long)cm0 * sc_m0 + cColOff;
          Cre[off] = cre[mi][ni][i];
          Cim[off] = cim[mi][ni][i];
        }
      }
    }
  }
}

// ---------------------------------------------------------------------------
// Cayley system build: per site, X = psi reshaped (256,128); A_skew = C - C^H
// with C = [X | 0].  M = I - A_skew (256x256), R = (I + A_skew)[:, :128].
// ---------------------------------------------------------------------------
__global__ void build_cayley(const float* __restrict__ psiRe,
                             const float* __restrict__ psiIm,
                             float* __restrict__ cay)
{
  const long idx = (long)blockIdx.x * blockDim.x + threadIdx.x;
  const long total = (long)NSITE * 256 * 256;
  if (idx >= total) return;
  const int site = (int)(idx >> 16);
  const int rem  = (int)(idx & 0xFFFF);
  const int i = rem >> 8;
  const int j = rem & 0xFF;

  const float* Xr = psiRe + (long)site * 32768;
  const float* Xi = psiIm + (long)site * 32768;

  float ar = 0.f, ai = 0.f;                  // A_skew(i,j)
  if (j < 128) {
    ar = Xr[i * 128 + j];
    ai = Xi[i * 128 + j];
    if (i < 128) { ar -= Xr[j * 128 + i]; ai += Xi[j * 128 + i]; }
  } else if (i < 128) {
    ar = -Xr[j * 128 + i];
    ai =  Xi[j * 128 + i];
  }

  float* Mre = cay + (long)site * 196608;
  float* Mim = Mre + 65536;
  float* Rre = Mre + 131072;
  float* Rim = Mre + 163840;
  const float dlt = (i == j) ? 1.f : 0.f;
  Mre[i * 256 + j] = dlt - ar;
  Mim[i * 256 + j] = -ai;
  if (j < 128) {
    Rre[i * 128 + j] = dlt + ar;
    Rim[i * 128 + j] = ai;
  }
}

// ---------------------------------------------------------------------------
// Gauss-Jordan (no pivoting; I - skewH is well conditioned). One 256-thread
// workgroup per site; pivot row staged in LDS; thread t owns row t.
// ---------------------------------------------------------------------------
__global__ void gauss_jordan(float* __restrict__ cay,
                             float* __restrict__ AisoRe,
                             float* __restrict__ AisoIm)
{
  const int site = blockIdx.x;
  float* Mre = cay + (long)site * 196608;
  float* Mim = Mre + 65536;
  float* Rre = Mre + 131072;
  float* Rim = Mre + 163840;

  __shared__ float prRe[384];
  __shared__ float prIm[384];
  __shared__ float s_ivr, s_ivi;
  const int t = threadIdx.x;   // 0..255

  for (int k = 0; k < 256; ++k) {
    if (t == 0) {
      const float pr = Mre[k * 256 + k], pi = Mim[k * 256 + k];
      const float den = pr * pr + pi * pi;
      s_ivr = pr / den;
      s_ivi = -pi / den;
    }
    __syncthreads();
    const float ivr = s_ivr, ivi = s_ivi;
    for (int j = t; j < 384; j += 256) {
      float xr, xi;
      if (j < 256) { xr = Mre[k * 256 + j]; xi = Mim[k * 256 + j]; }
      else         { xr = Rre[k * 128 + (j - 256)]; xi = Rim[k * 128 + (j - 256)]; }
      const float yr = xr * ivr - xi * ivi;
      const float yi = xr * ivi + xi * ivr;
      prRe[j] = yr; prIm[j] = yi;
      if (j < 256) { Mre[k * 256 + j] = yr; Mim[k * 256 + j] = yi; }
      else         { Rre[k * 128 + (j - 256)] = yr; Rim[k * 128 + (j - 256)] = yi; }
    }
    __syncthreads();
    if (t != k) {
      const float fr = Mre[t * 256 + k], fi = Mim[t * 256 + k];
      for (int j = k; j < 256; ++j) {
        const float yr = prRe[j], yi = prIm[j];
        Mre[t * 256 + j] -= fr * yr - fi * yi;
        Mim[t * 256 + j] -= fr * yi + fi * yr;
      }
      for (int j = 0; j < 128; ++j) {
        const float yr = prRe[256 + j], yi = prIm[256 + j];
        Rre[t * 128 + j] -= fr * yr - fi * yi;
        Rim[t * 128 + j] -= fr * yi + fi * yr;
      }
    }
    __syncthreads();
  }
  for (int i = t; i < 32768; i += 256) {
    AisoRe[(long)site * 32768 + i] = Rre[i];
    AisoIm[(long)site * 32768 + i] = Rim[i];
  }
}

// ---------------------------------------------------------------------------
// W[(a,b,t),(u,d,c)] = sum_s conj(H[a,s,u,d]) * H[b,s,t,c]   (288 x 288)
// H site layout: (a, s, t, b) with strides (48, 24, 12, 1).
// ---------------------------------------------------------------------------
__global__ void build_w(const float* __restrict__ Hre,
                        const float* __restrict__ Him,
                        float* __restrict__ Wre, float* __restrict__ Wim)
{
  const int idx = blockIdx.x * blockDim.x + threadIdx.x;
  if (idx >= 288 * 288) return;
  const int row = idx / 288, col = idx - row * 288;
  const int a = row / 24, b = (row % 24) >> 1, tt = row & 1;
  const int u = col / 144, d = (col % 144) / 12, c = col % 12;
  float accR = 0.f, accI = 0.f;
#pragma unroll
  for (int s = 0; s < 2; ++s) {
    const int o1 = a * 48 + s * 24 + u * 12 + d;   // H[a,s,u,d]  (conjugated)
    const int o2 = b * 48 + s * 24 + tt * 12 + c;  // H[b,s,t,c]
    const float xr = Hre[o1], xi = Him[o1];
    const float yr = Hre[o2], yi = Him[o2];
    accR += xr * yr + xi * yi;
    accI += xr * yi - xi * yr;
  }
  Wre[idx] = accR;
  Wim[idx] = accI;
}

// env initialization: zero both planes, re[0] = 1
__global__ void init_env(float* __restrict__ re, float* __restrict__ im, int n)
{
  const int i = blockIdx.x * blockDim.x + threadIdx.x;
  if (i < n) {
    re[i] = (i == 0) ? 1.f : 0.f;
    im[i] = 0.f;
  }
}

// val = expHH - E*expH - conj(E*expH) + |E|^2 ;  out = [val.re, val.im]
__global__ void finalize(const float* __restrict__ eHre, const float* __restrict__ eHim,
                         const float* __restrict__ eHHre, const float* __restrict__ eHHim,
                         const float* __restrict__ Ere, const float* __restrict__ Eim,
                         float* __restrict__ out)
{
  if (threadIdx.x == 0 && blockIdx.x == 0) {
    const float hr = eHre[0],  hi = eHim[0];
    const float hhr = eHHre[0], hhi = eHHim[0];
    const float er = Ere[0], ei = Eim[0];
    const float br = er * hr - ei * hi;          // Re(E*expH)
    out[0] = hhr - 2.f * br + (er * er + ei * ei);
    out[1] = hhi;                                // imag parts of B cancel
  }
}

// ---------------------------------------------------------------------------
// Host side
// ---------------------------------------------------------------------------
static void launch_cgemm(hipStream_t s,
                         int M1, int M0, int N1, int N0, int K,
                         const float* Are, const float* Aim,
                         long sam1, long sam0, long sak1, long sak0, int K0,
                         const float* Bre, const float* Bim,
                         long sbk1, long sbk0, long sbn1, long sbn0,
                         float* Cre, float* Cim,
                         long scm1, long scm0, long scn1, long scn0,
                         int conjA)
{
  const int M = M1 * M0, N = N1 * N0;
  const long adjA = sak1 - (long)K0 * sak0;
  const long adjB = sbk1 - (long)K0 * sbk0;
  long dA = 4 * sak0, dB = 4 * sbk0;
  int K0eff = K0;
  if (K0 < 4) {                 // K0 divides the step: constant wrap count
    const int wraps = 4 / K0;   // (all calls use K0 in {2,128,288})
    dA += (long)wraps * adjA;
    dB += (long)wraps * adjB;
    K0eff = 0x7fffffff;         // disable in-loop wrap check
  }
  dim3 grid((N + 63) / 64, (M + 31) / 32);
  if (conjA)
    cgemm_wmma<1><<<grid, 32, 0, s>>>(M1, M0, N1, N0, K,
                                      Are, Aim, sam1, sam0, sak0, adjA, dA,
                                      Bre, Bim, sbk0, adjB, dB, sbn1, sbn0,
                                      K0, K0eff,
                                      Cre, Cim, scm1, scm0, scn1, scn0);
  else
    cgemm_wmma<0><<<grid, 32, 0, s>>>(M1, M0, N1, N0, K,
                                      Are, Aim, sam1, sam0, sak0, adjA, dA,
                                      Bre, Bim, sbk0, adjB, dB, sbn1, sbn0,
                                      K0, K0eff,
                                      Cre, Cim, scm1, scm0, scn1, scn0);
}

extern "C" void kernel_launch(void* const* d_in, const int* in_sizes, int n_in,
                              void* d_out, int out_size, void* d_ws, size_t ws_size,
                              hipStream_t stream)
{
  (void)in_sizes; (void)n_in; (void)out_size; (void)ws_size;
  const float* psiRe = (const float*)d_in[0];  // (32,128,2,128)
  const float* psiIm = (const float*)d_in[1];
  const float* HRe   = (const float*)d_in[2];  // (32,12,2,2,12)
  const float* HIm   = (const float*)d_in[3];
  const float* ERe   = (const float*)d_in[4];
  const float* EIm   = (const float*)d_in[5];
  float* out = (float*)d_out;

  // workspace bump allocation (floats); total ~132 MB
  float* w = (float*)d_ws;
  size_t o = 0;
  float* AisoRe = w + o; o += (size_t)NSITE * 32768;
  float* AisoIm = w + o; o += (size_t)NSITE * 32768;
  float* envHaRe = w + o; o += 196608;   // 128*12*128
  float* envHaIm = w + o; o += 196608;
  float* envHbRe = w + o; o += 196608;
  float* envHbIm = w + o; o += 196608;
  float* envHHaRe = w + o; o += 2359296; // 128*12*12*128
  float* envHHaIm = w + o; o += 2359296;
  float* envHHbRe = w + o; o += 2359296;
  float* envHHbIm = w + o; o += 2359296;
  float* Wre = w + o; o += 82944;        // 288*288
  float* Wim = w + o; o += 82944;
  float* T1hRe = w + o; o += 393216;     // 128*12*2*128
  float* T1hIm = w + o; o += 393216;
  float* T2hRe = w + o; o += 393216;     // 128*2*12*128
  float* T2hIm = w + o; o += 393216;
  float* T1hhRe = w + o; o += 4718592;   // 128*144*2*128
  float* T1hhIm = w + o; o += 4718592;
  float* T3hhRe = w + o; o += 4718592;   // 128*128*288
  float* T3hhIm = w + o; o += 4718592;
  // Cayley scratch (32 * 256*384 complex = 6.29M floats) reuses the T1hh
  // region (9.4M floats): the Cayley phase completes before any hh GEMM runs.
  float* cay = T1hhRe;

  // Phase 1: Cayley isometrization of all 32 sites
  {
    const long total = (long)NSITE * 256 * 256;
    build_cayley<<<(unsigned)((total + 255) / 256), 256, 0, stream>>>(psiRe, psiIm, cay);
    gauss_jordan<<<NSITE, 256, 0, stream>>>(cay, AisoRe, AisoIm);
  }

  // Phase 2: environment scans
  init_env<<<(196608 + 255) / 256, 256, 0, stream>>>(envHaRe, envHaIm, 196608);
  init_env<<<(2359296 + 255) / 256, 256, 0, stream>>>(envHHaRe, envHHaIm, 2359296);

  float* eHaR = envHaRe;  float* eHaI = envHaIm;
  float* eHbR = envHbRe;  float* eHbI = envHbIm;
  float* eHHaR = envHHaRe; float* eHHaI = envHHaIm;
  float* eHHbR = envHHbRe; float* eHHbI = envHHbIm;

  for (int site = 0; site < NSITE; ++site) {
    const float* AsR = AisoRe + (long)site * 32768;  // (m,t,r): m*256+t*128+r
    const float* AsI = AisoIm + (long)site * 32768;
    const float* HsR = HRe + (long)site * 576;       // (a,s,t,b): a*48+s*24+t*12+b
    const float* HsI = HIm + (long)site * 576;

    // ---- <H> scan ----
    // s1: T1[l,a,t,r] = sum_m env[l,a,m] * A[m,t,r]
    launch_cgemm(stream, 128, 12, 2, 128, 128,
                 eHaR, eHaI, 1536, 128, 0, 1, 128,
                 AsR, AsI, 0, 256, 128, 1,
                 T1hRe, T1hIm, 3072, 256, 128, 1, 0);
    // s2: T2[l,s,b,r] = sum_{a,t} T1[l,a,t,r] * H[a,s,t,b]
    launch_cgemm(stream, 128, 128, 2, 12, 24,
                 T1hRe, T1hIm, 3072, 1, 256, 128, 2,
                 HsR, HsI, 48, 12, 24, 1,
                 T2hRe, T2hIm, 3072, 1, 1536, 128, 0);
    // s3: env'[q,b,r] = sum_{l,s} conj(A[l,s,q]) * T2[l,s,b,r]
    launch_cgemm(stream, 1, 128, 12, 128, 256,
                 AsR, AsI, 0, 1, 256, 128, 2,
                 T2hRe, T2hIm, 3072, 1536, 128, 1,
                 eHbR, eHbI, 0, 1536, 128, 1, 1);

    // ---- <H^dag H> scan ----
    build_w<<<(288 * 288 + 255) / 256, 256, 0, stream>>>(HsR, HsI, Wre, Wim);
    // hh1: T1[l,(a,b),t,r] = sum_m env[l,a,b,m] * A[m,t,r]
    launch_cgemm(stream, 128, 144, 2, 128, 128,
                 eHHaR, eHHaI, 18432, 128, 0, 1, 128,
                 AsR, AsI, 0, 256, 128, 1,
                 T1hhRe, T1hhIm, 36864, 256, 128, 1, 0);
    // hh2: T3[l,r,(u,d,c)] = sum_{abt} T1[l,(abt),r] * W[(abt),(udc)]
    launch_cgemm(stream, 128, 128, 1, 288, 288,
                 T1hhRe, T1hhIm, 36864, 1, 0, 128, 288,
                 Wre, Wim, 0, 288, 0, 1,
                 T3hhRe, T3hhIm, 36864, 288, 0, 1, 0);
    // hh3: env'[q,d,c,r] = sum_{l,u} conj(A[l,u,q]) * T3[l,r,(u*144+dc)]
    launch_cgemm(stream, 1, 128, 128, 144, 256,
                 AsR, AsI, 0, 1, 256, 128, 2,
                 T3hhRe, T3hhIm, 36864, 144, 288, 1,
                 eHHbR, eHHbI, 0, 18432, 1, 128, 1);

    // ping-pong envs
    float* t;
    t = eHaR; eHaR = eHbR; eHbR = t;
    t = eHaI; eHaI = eHbI; eHbI = t;
    t = eHHaR; eHHaR = eHHbR; eHHbR = t;
    t = eHHaI; eHHaI = eHHbI; eHHbI = t;
  }

  finalize<<<1, 32, 0, stream>>>(eHaR, eHaI, eHHaR, eHHaI, ERe, EIm, out);
}